// MLA_81209241633268
// MI455X (gfx1250) — compile-verified
//
#include <hip/hip_runtime.h>
#include <math.h>

// ---------------- problem constants ----------------
#define DIMM      2048
#define N_HEADS   16
#define Q_LORA    1536
#define KV_LORA   512
#define QK_NOPE   128
#define QK_ROPE   64
#define V_DIM     128
#define QK_HD     (QK_NOPE + QK_ROPE)   // 192
#define MAX_SEQ   8192
#define BSZ       32
#define SEQLEN    4
#define START_POS (MAX_SEQ - SEQLEN)    // 8188
#define NTOK      (BSZ * SEQLEN)        // 128
#define RQ        (SEQLEN * N_HEADS)    // 64 query rows per batch
#define DQ        (KV_LORA + QK_ROPE)   // 576 latent dim (kv | pe)

// LDS strides (floats). KS_LD = 512 + 2x2-dword TDM pads (interval 256dw).
// PE_LD = 64 + 4-dword TDM pad (interval 64dw). All 16B-aligned rows,
// row-step = 4 banks -> 16 lanes hit 16 distinct banks.
#define QS_LD 580
#define KS_LD 516
#define PE_LD 68
#define PS_LD 68

typedef __attribute__((ext_vector_type(2))) float    v2f;
typedef __attribute__((ext_vector_type(8))) float    v8f;
typedef __attribute__((ext_vector_type(4))) unsigned v4u;
typedef __attribute__((ext_vector_type(4))) int      v4i;
typedef __attribute__((ext_vector_type(8))) int      v8i;

#if defined(__has_builtin)
# if __has_builtin(__builtin_amdgcn_tensor_load_to_lds)
#  define HAVE_TDM 1
# else
#  define HAVE_TDM 0
# endif
#else
# define HAVE_TDM 0
#endif

#if defined(__has_include)
# if __has_include(<hip/amd_detail/amd_gfx1250_TDM.h>)
#  define TDM_SIX_ARGS 1
# else
#  define TDM_SIX_ARGS 0
# endif
#else
# define TDM_SIX_ARGS 0
#endif

// D(16x16) = A(16x4) * B(4x16) + C, fp32 native WMMA (CDNA5)
static __device__ inline v8f wmma4(v2f a, v2f b, v8f c) {
    return __builtin_amdgcn_wmma_f32_16x16x4_f32(
        false, a, false, b, (short)0, c, false, false);
}

// Async global -> LDS copy (CDNA5, ASYNCcnt-tracked)
static __device__ inline void async_b128(unsigned lds_off, const void* gsrc) {
    asm volatile("global_load_async_to_lds_b128 %0, %1, off"
                 :: "v"(lds_off), "v"(gsrc) : "memory");
}
static __device__ inline void wait_async0() {
    asm volatile("s_wait_asynccnt 0" ::: "memory");
}
static __device__ inline unsigned lds_addr(const void* p) {
    return (unsigned)(unsigned long long)p;   // low 32 bits = LDS offset
}

#if HAVE_TDM
// TDM: DMA a 2-D fp32 tile (rows x row_dwords) global -> LDS with the
// pad unit inserting LDS padding. pad codes per ISA: interval 2^(c+1)
// dwords, amount (c+1) dwords.  Issue once per wave; TENSORcnt-tracked.
static __device__ inline void tdm_load_2d(
    unsigned lds_byte, const void* gaddr, unsigned rows,
    unsigned row_dwords, unsigned pad_int_c, unsigned pad_amt_c)
{
    unsigned long long ga = (unsigned long long)gaddr;
    v4u g0;
    g0[0] = 1u;                                         // count=1 (user D#)
    g0[1] = lds_byte;                                   // lds_addr
    g0[2] = (unsigned)(ga & 0xFFFFFFFFu);               // global_addr lo
    g0[3] = (unsigned)((ga >> 32) & 0x01FFFFFFu) | (2u << 30);  // hi | type=2
    v8i g1;
    g1[0] = (int)((2u << 16) | (1u << 20) | (pad_int_c << 22) | (pad_amt_c << 25));
    g1[1] = (int)(row_dwords << 16);                    // tensor_dim0 lo16
    g1[2] = (int)(rows << 16);                          // tensor_dim1 lo16
    g1[3] = (int)(row_dwords << 16);                    // tile_dim0
    g1[4] = (int)rows;                                  // tile_dim1
    g1[5] = (int)row_dwords;                            // tensor_dim0_stride
    g1[6] = 0;
    g1[7] = 0;
    v4i z4 = {0, 0, 0, 0};
#if TDM_SIX_ARGS
    v8i z8 = {0, 0, 0, 0, 0, 0, 0, 0};
    __builtin_amdgcn_tensor_load_to_lds(g0, g1, z4, z4, z8, 0);
#else
    __builtin_amdgcn_tensor_load_to_lds(g0, g1, z4, z4, 0);
#endif
}
#endif

// =====================================================================
// Generic batched WMMA GEMM: C = alpha * A(MxK) * B(KxN), row-major.
// Block tile 64x64, 256 threads (8 waves), K staged 32-wide via
// global_load_async_to_lds_b128.  grid = (N/64, M/64, batch)
// =====================================================================
__global__ __launch_bounds__(256) void wmma_gemm(
    const float* __restrict__ A, int lda, long long sA,
    const float* __restrict__ B, int ldb, long long sB,
    float*       __restrict__ C, int ldc, long long sC,
    int M, int N, int K, float alpha)
{
    __shared__ __attribute__((aligned(16))) float As[64 * 36];
    __shared__ __attribute__((aligned(16))) float Bs[32 * 68];

    const int tid  = threadIdx.x;
    const int lane = tid & 31;
    const int wave = tid >> 5;
    const int half = lane >> 4;
    const int l15  = lane & 15;
    const int r    = wave >> 1;
    const int cp   = wave & 1;

    const float* Ab = A + (long long)blockIdx.z * sA + (long long)blockIdx.y * 64 * lda;
    const float* Bb = B + (long long)blockIdx.z * sB + blockIdx.x * 64;
    float*       Cb = C + (long long)blockIdx.z * sC + (long long)blockIdx.y * 64 * ldc
                        + blockIdx.x * 64;

    v8f acc0 = {0.f,0.f,0.f,0.f,0.f,0.f,0.f,0.f};
    v8f acc1 = acc0;

    for (int k0 = 0; k0 < K; k0 += 32) {
        __syncthreads();
        #pragma unroll
        for (int k = 0; k < 2; k++) {
            int i = tid + k * 256;
            int row = i >> 3, c4 = i & 7;
            async_b128(lds_addr(&As[row * 36 + c4 * 4]),
                       Ab + (long long)row * lda + k0 + c4 * 4);
        }
        #pragma unroll
        for (int k = 0; k < 2; k++) {
            int i = tid + k * 256;
            int row = i >> 4, n4 = i & 15;
            async_b128(lds_addr(&Bs[row * 68 + n4 * 4]),
                       Bb + (long long)(k0 + row) * ldb + n4 * 4);
        }
        wait_async0();
        __syncthreads();
        #pragma unroll
        for (int ks = 0; ks < 8; ks++) {
            const int kb = ks * 4 + 2 * half;
            v2f a = *(const v2f*)&As[(r * 16 + l15) * 36 + kb];
            v2f b0, b1;
            b0[0] = Bs[kb       * 68 + (cp * 2) * 16 + l15];
            b0[1] = Bs[(kb + 1) * 68 + (cp * 2) * 16 + l15];
            b1[0] = Bs[kb       * 68 + (cp * 2 + 1) * 16 + l15];
            b1[1] = Bs[(kb + 1) * 68 + (cp * 2 + 1) * 16 + l15];
            acc0 = wmma4(a, b0, acc0);
            acc1 = wmma4(a, b1, acc1);
        }
    }
    #pragma unroll
    for (int j = 0; j < 8; j++) {
        int row = r * 16 + j + 8 * half;
        Cb[(long long)row * ldc + (cp * 2) * 16 + l15]     = alpha * acc0[j];
        Cb[(long long)row * ldc + (cp * 2 + 1) * 16 + l15] = alpha * acc1[j];
    }
}

// ===== RMSNorm(x@wq_a) -> qn ; RMSNorm+RoPE(x@wkv_a) -> newkv =====
__global__ __launch_bounds__(256) void mla_norm_rope(
    const float* __restrict__ xqa, const float* __restrict__ qnw,
    float* __restrict__ qn,
    const float* __restrict__ kvfull, const float* __restrict__ kvnw,
    float* __restrict__ newkv,
    const float* __restrict__ fcos, const float* __restrict__ fsin)
{
    const int tok = blockIdx.x, tid = threadIdx.x;
    __shared__ float red[256];

    float ss = 0.f;
    for (int i = tid; i < Q_LORA; i += 256) { float v = xqa[tok * Q_LORA + i]; ss += v * v; }
    red[tid] = ss; __syncthreads();
    for (int off = 128; off; off >>= 1) { if (tid < off) red[tid] += red[tid + off]; __syncthreads(); }
    const float rq = rsqrtf(red[0] / (float)Q_LORA + 1e-6f);
    for (int i = tid; i < Q_LORA; i += 256)
        qn[tok * Q_LORA + i] = xqa[tok * Q_LORA + i] * rq * qnw[i];

    float ss2 = 0.f;
    for (int i = tid; i < KV_LORA; i += 256) { float v = kvfull[tok * DQ + i]; ss2 += v * v; }
    __syncthreads();
    red[tid] = ss2; __syncthreads();
    for (int off = 128; off; off >>= 1) { if (tid < off) red[tid] += red[tid + off]; __syncthreads(); }
    const float rk = rsqrtf(red[0] / (float)KV_LORA + 1e-6f);
    for (int i = tid; i < KV_LORA; i += 256)
        newkv[tok * DQ + i] = kvfull[tok * DQ + i] * rk * kvnw[i];

    if (tid < QK_ROPE / 2) {
        const int s = tok & 3;
        float x0 = kvfull[tok * DQ + KV_LORA + 2 * tid];
        float x1 = kvfull[tok * DQ + KV_LORA + 2 * tid + 1];
        float c = fcos[s * 32 + tid], sn = fsin[s * 32 + tid];
        newkv[tok * DQ + KV_LORA + 2 * tid]     = x0 * c - x1 * sn;
        newkv[tok * DQ + KV_LORA + 2 * tid + 1] = x0 * sn + x1 * c;
    }
}

// ===== RoPE(q_pe) -> scaled pe columns of Qws[b][s*16+h][512..575] =====
__global__ __launch_bounds__(512) void mla_qpe(
    const float* __restrict__ q, const float* __restrict__ fcos,
    const float* __restrict__ fsin, float* __restrict__ Qws, float scale)
{
    const int tok = blockIdx.x, tid = threadIdx.x;
    const int b = tok >> 2, s = tok & 3;
    const int h = tid >> 5, p = tid & 31;
    float x0 = q[tok * (N_HEADS * QK_HD) + h * QK_HD + QK_NOPE + 2 * p];
    float x1 = q[tok * (N_HEADS * QK_HD) + h * QK_HD + QK_NOPE + 2 * p + 1];
    float c = fcos[s * 32 + p], sn = fsin[s * 32 + p];
    float* dst = Qws + (long long)b * (RQ * DQ) + (s * N_HEADS + h) * DQ + KV_LORA;
    dst[2 * p]     = scale * (x0 * c - x1 * sn);
    dst[2 * p + 1] = scale * (x0 * sn + x1 * c);
}

// ===== transpose Wv: WvT[h][c][d] = wkv_b[(h*256+128+d)*512 + c] =====
__global__ __launch_bounds__(256) void mla_wvt(
    const float* __restrict__ wkv_b, float* __restrict__ WvT)
{
    int idx = blockIdx.x * 256 + threadIdx.x;
    int h = idx >> 16, c = (idx >> 7) & 511, d = idx & 127;
    WvT[idx] = wkv_b[((h * 256 + 128) + d) * KV_LORA + c];
}

// =====================================================================
// Flash-attention chunk kernel. grid = (nchunk, BSZ), 512 threads
// (16 waves). K/PE tiles DMA'd by the Tensor Data Mover (TENSORcnt),
// Q staged via global_load_async_to_lds_b128 (ASYNCcnt).
// =====================================================================
__global__ __launch_bounds__(512) void mla_flash_chunk(
    const float* __restrict__ Qws,        // [BSZ][RQ][DQ], pre-scaled
    const float* __restrict__ kv_cache,   // [BSZ][MAX_SEQ][KV_LORA]
    const float* __restrict__ pe_cache,   // [BSZ][MAX_SEQ][QK_ROPE]
    const float* __restrict__ newkv,      // [NTOK][DQ]
    float* __restrict__ pO,               // [BSZ][nchunk][RQ][KV_LORA]
    float* __restrict__ pml,              // [BSZ][nchunk][RQ][2]
    int nchunk, int chlen)
{
    extern __shared__ __attribute__((aligned(16))) float sm[];
    float* Qs   = sm;                       // 64*580  (kv|pe per row)
    float* Ks   = Qs + 64 * QS_LD;          // 64*516  (kv, TDM-padded)
    float* Pe   = Ks + 64 * KS_LD;          // 64*68   (pe, TDM-padded)
    float* Ps   = Pe + 64 * PE_LD;          // 64*68   (scores/probs)
    float* mrow = Ps + 64 * PS_LD;
    float* lrow = mrow + 64;
    float* arow = lrow + 64;

    const int tid  = threadIdx.x;
    const int lane = tid & 31;
    const int wave = tid >> 5;
    const int half = lane >> 4;
    const int l15  = lane & 15;
    const int r    = wave >> 2;             // query row tile 0..3
    const int ct   = wave & 3;              // score t-tile
    const int cg   = wave & 3;              // output col group (128 wide)
    const int b    = blockIdx.y;
    const int ch   = blockIdx.x;
    const int koff = (cg >= 2) ? 2 : 0;     // mid-row TDM pad skip (c>=256)

    // ---- async-stage Q [64 x 576] (one time) ----
    for (int idx = tid; idx < 64 * (DQ / 4); idx += 512) {
        int row = idx / (DQ / 4), c4 = idx - row * (DQ / 4);
        async_b128(lds_addr(&Qs[row * QS_LD + c4 * 4]),
                   Qws + (long long)b * (RQ * DQ) + row * DQ + c4 * 4);
    }
    if (tid < RQ) { mrow[tid] = -3.0e38f; lrow[tid] = 0.f; }

    v8f o[8];
    #pragma unroll
    for (int u = 0; u < 8; u++) { v8f z = {0.f,0.f,0.f,0.f,0.f,0.f,0.f,0.f}; o[u] = z; }

    const int nblk = chlen >> 6;
    for (int tb = 0; tb < nblk; tb++) {
        const int t0 = ch * chlen + tb * 64;
        int rows = START_POS - t0; if (rows > 64) rows = 64;   // cache rows
        int tail = 64 - rows;                                  // ws rows (<=4)
        __syncthreads();                      // prior reads of Ks/Pe done

#if HAVE_TDM
        if (tid == 0) {
            tdm_load_2d(lds_addr(Ks), &kv_cache[((long long)b * MAX_SEQ + t0) * KV_LORA],
                        (unsigned)rows, KV_LORA, 7u, 1u);      // stride 516
            tdm_load_2d(lds_addr(Pe), &pe_cache[((long long)b * MAX_SEQ + t0) * QK_ROPE],
                        (unsigned)rows, QK_ROPE, 5u, 3u);      // stride 68
        }
#else
        #pragma unroll
        for (int k = 0; k < 16; k++) {
            int i = tid + k * 512;
            int row = i >> 7, c4 = i & 127;
            if (row < rows)
                async_b128(lds_addr(&Ks[row * KS_LD + c4 * 4 + ((c4 >= 64) ? 2 : 0)]),
                           &kv_cache[((long long)b * MAX_SEQ + t0 + row) * KV_LORA + c4 * 4]);
        }
        #pragma unroll
        for (int k = 0; k < 2; k++) {
            int i = tid + k * 512;
            int row = i >> 4, c4 = i & 15;
            if (row < rows)
                async_b128(lds_addr(&Pe[row * PE_LD + c4 * 4]),
                           &pe_cache[((long long)b * MAX_SEQ + t0 + row) * QK_ROPE + c4 * 4]);
        }
#endif
        // ---- tail rows (updated cache entries) from workspace ----
        for (int idx = tid; idx < tail * DQ; idx += 512) {
            int rr = idx / DQ, cc = idx - rr * DQ;
            int row = rows + rr;
            float v = newkv[(b * SEQLEN + (t0 + row - START_POS)) * DQ + cc];
            if (cc < KV_LORA) Ks[row * KS_LD + cc + ((cc >= 256) ? 2 : 0)] = v;
            else              Pe[row * PE_LD + (cc - KV_LORA)] = v;
        }
        // prefetch next block's kv rows (global_prefetch_b8)
        if (tid < 64) {
            int tn = t0 + 64 + tid;
            if ((tb + 1 < nblk) && (tn < START_POS))
                __builtin_prefetch(&kv_cache[((long long)b * MAX_SEQ + tn) * KV_LORA], 0, 1);
        }
        wait_async0();
#if HAVE_TDM
        if (tid == 0) __builtin_amdgcn_s_wait_tensorcnt(0);
#endif
        __syncthreads();

        // ---- S = Q*K^T : 64+64 steps over kv (pad split) + 16 over pe ----
        v8f s = {0.f,0.f,0.f,0.f,0.f,0.f,0.f,0.f};
        {
            const float* qrow = &Qs[(r * 16 + l15) * QS_LD];
            const float* krow = &Ks[(ct * 16 + l15) * KS_LD];
            for (int kk = 0; kk < 64; kk++) {
                const int kb = kk * 4 + 2 * half;
                v2f a  = *(const v2f*)&qrow[kb];
                v2f bb = *(const v2f*)&krow[kb];
                s = wmma4(a, bb, s);
            }
            for (int kk = 64; kk < 128; kk++) {
                const int kb = kk * 4 + 2 * half;
                v2f a  = *(const v2f*)&qrow[kb];
                v2f bb = *(const v2f*)&krow[kb + 2];   // skip mid-row pad
                s = wmma4(a, bb, s);
            }
            const float* qpe = qrow + KV_LORA;
            const float* kpe = &Pe[(ct * 16 + l15) * PE_LD];
            #pragma unroll
            for (int kk = 0; kk < QK_ROPE / 4; kk++) {
                const int kb = kk * 4 + 2 * half;
                v2f a  = *(const v2f*)&qpe[kb];
                v2f bb = *(const v2f*)&kpe[kb];
                s = wmma4(a, bb, s);
            }
        }
        #pragma unroll
        for (int j = 0; j < 8; j++)
            Ps[(r * 16 + j + 8 * half) * PS_LD + ct * 16 + l15] = s[j];
        __syncthreads();

        // ---- online softmax per row ----
        if (tid < RQ) {
            float mo = mrow[tid], mx = mo;
            for (int t = 0; t < 64; t++) mx = fmaxf(mx, Ps[tid * PS_LD + t]);
            float a = __expf(mo - mx);
            float ls = 0.f;
            for (int t = 0; t < 64; t++) {
                float p = __expf(Ps[tid * PS_LD + t] - mx);
                Ps[tid * PS_LD + t] = p; ls += p;
            }
            lrow[tid] = lrow[tid] * a + ls;
            mrow[tid] = mx;
            arow[tid] = a;
        }
        __syncthreads();

        // ---- O = O*alpha + P * V ----
        float af[8];
        #pragma unroll
        for (int j = 0; j < 8; j++) af[j] = arow[r * 16 + j + 8 * half];
        #pragma unroll
        for (int u = 0; u < 8; u++) {
            #pragma unroll
            for (int j = 0; j < 8; j++) o[u][j] *= af[j];
        }
        for (int kk = 0; kk < 16; kk++) {
            const int kb = kk * 4 + 2 * half;
            v2f a = *(const v2f*)&Ps[(r * 16 + l15) * PS_LD + kb];
            #pragma unroll
            for (int u = 0; u < 8; u++) {
                const int col = cg * 128 + u * 16 + l15 + koff;
                v2f bb;
                bb[0] = Ks[kb * KS_LD + col];
                bb[1] = Ks[(kb + 1) * KS_LD + col];
                o[u] = wmma4(a, bb, o[u]);
            }
        }
    }

    // ---- write partials ----
    const long long base = ((long long)(b * nchunk + ch)) * RQ * KV_LORA;
    #pragma unroll
    for (int u = 0; u < 8; u++) {
        #pragma unroll
        for (int j = 0; j < 8; j++) {
            int row = r * 16 + j + 8 * half;
            int col = cg * 128 + u * 16 + l15;
            pO[base + (long long)row * KV_LORA + col] = o[u][j];
        }
    }
    __syncthreads();
    if (tid < RQ) {
        pml[(b * nchunk + ch) * (RQ * 2) + tid * 2]     = mrow[tid];
        pml[(b * nchunk + ch) * (RQ * 2) + tid * 2 + 1] = lrow[tid];
    }
}

// ===== combine chunk partials (log-sum-exp) -> attn[b][s][h][512] =====
__global__ __launch_bounds__(128) void mla_reduce(
    const float* __restrict__ pO, const float* __restrict__ pml,
    float* __restrict__ attn, int nchunk)
{
    const int b = blockIdx.x >> 6, row = blockIdx.x & 63;
    const int s = row >> 4, h = row & 15;
    float mstar = -3.0e38f;
    for (int j = 0; j < nchunk; j++)
        mstar = fmaxf(mstar, pml[(b * nchunk + j) * (RQ * 2) + row * 2]);
    float w[8];
    float lsum = 0.f;
    for (int j = 0; j < nchunk; j++) {
        float wj = __expf(pml[(b * nchunk + j) * (RQ * 2) + row * 2] - mstar);
        w[j] = wj;
        lsum += wj * pml[(b * nchunk + j) * (RQ * 2) + row * 2 + 1];
    }
    const float inv = 1.f / lsum;
    for (int c = threadIdx.x; c < KV_LORA; c += 128) {
        float acc = 0.f;
        for (int j = 0; j < nchunk; j++)
            acc += w[j] * pO[(((long long)(b * nchunk + j)) * RQ + row) * KV_LORA + c];
        attn[(long long)b * (SEQLEN * 8192) + s * 8192 + h * KV_LORA + c] = acc * inv;
    }
}

// =====================================================================
extern "C" void kernel_launch(void* const* d_in, const int* in_sizes, int n_in,
                              void* d_out, int out_size, void* d_ws, size_t ws_size,
                              hipStream_t stream)
{
    (void)in_sizes; (void)n_in; (void)out_size;
    const float* x        = (const float*)d_in[0];
    const float* wq_a     = (const float*)d_in[1];
    const float* q_norm_w = (const float*)d_in[2];
    const float* wq_b     = (const float*)d_in[3];
    const float* wkv_a    = (const float*)d_in[4];
    const float* kv_norm_w= (const float*)d_in[5];
    const float* wkv_b    = (const float*)d_in[6];
    const float* wo       = (const float*)d_in[7];
    const float* kv_cache = (const float*)d_in[8];
    const float* pe_cache = (const float*)d_in[9];
    const float* fcos     = (const float*)d_in[10];
    const float* fsin     = (const float*)d_in[11];
    float* out = (float*)d_out;

    float* ws = (float*)d_ws;
    float* ws_xqa    = ws;                 // 128*1536
    float* ws_qn     = ws + 196608;        // 128*1536
    float* ws_kvfull = ws + 393216;        // 128*576
    float* ws_newkv  = ws + 466944;        // 128*576
    float* ws_q      = ws + 540672;        // 128*3072
    float* ws_Q      = ws + 933888;        // 32*64*576
    float* ws_WvT    = ws + 2113536;       // 16*512*128
    float* ws_attn   = ws + 3162112;       // 128*16*512
    float* ws_out2   = ws + 4210688;       // 128*2048
    float* ws_pml    = ws + 4472832;       // 32*8*64*2
    float* ws_pO     = ws + 4505600;       // 32*nchunk*64*512

    int nchunk = 8;
    while (nchunk > 1 &&
           (4505600ULL + (size_t)nchunk * 1048576ULL) * 4ULL > ws_size)
        nchunk >>= 1;
    const int chlen = MAX_SEQ / nchunk;

    const float scale = 1.0f / sqrtf((float)QK_HD);
    const size_t fa_lds =
        (size_t)(64 * QS_LD + 64 * KS_LD + 64 * PE_LD + 64 * PS_LD + 192) * sizeof(float);
    hipFuncSetAttribute((const void*)mla_flash_chunk,
                        hipFuncAttributeMaxDynamicSharedMemorySize, (int)fa_lds);

    // 1) xqa = x @ wq_a            (128 x 2048 x 1536)
    wmma_gemm<<<dim3(Q_LORA/64, NTOK/64, 1), 256, 0, stream>>>(
        x, DIMM, 0, wq_a, Q_LORA, 0, ws_xqa, Q_LORA, 0, NTOK, Q_LORA, DIMM, 1.0f);
    // 2) kvfull = x @ wkv_a        (128 x 2048 x 576)
    wmma_gemm<<<dim3(DQ/64, NTOK/64, 1), 256, 0, stream>>>(
        x, DIMM, 0, wkv_a, DQ, 0, ws_kvfull, DQ, 0, NTOK, DQ, DIMM, 1.0f);
    // 3) rmsnorm(q), rmsnorm(kv)+rope(k_pe)
    mla_norm_rope<<<NTOK, 256, 0, stream>>>(
        ws_xqa, q_norm_w, ws_qn, ws_kvfull, kv_norm_w, ws_newkv, fcos, fsin);
    // 4) q = qn @ wq_b             (128 x 1536 x 3072)
    wmma_gemm<<<dim3((N_HEADS*QK_HD)/64, NTOK/64, 1), 256, 0, stream>>>(
        ws_qn, Q_LORA, 0, wq_b, N_HEADS*QK_HD, 0, ws_q, N_HEADS*QK_HD, 0,
        NTOK, N_HEADS*QK_HD, Q_LORA, 1.0f);
    // 5) rope(q_pe) -> scaled pe columns of Qws
    mla_qpe<<<NTOK, 512, 0, stream>>>(ws_q, fcos, fsin, ws_Q, scale);
    // 6) absorbed q_nope: per-head (128 x 128 x 512), alpha=scale
    wmma_gemm<<<dim3(KV_LORA/64, NTOK/64, N_HEADS), 256, 0, stream>>>(
        ws_q, N_HEADS*QK_HD, QK_HD,
        wkv_b, KV_LORA, (long long)(QK_NOPE+V_DIM)*KV_LORA,
        ws_Q, 9216, DQ,
        NTOK, KV_LORA, QK_NOPE, scale);
    // 7) transpose Wv
    mla_wvt<<<(N_HEADS*KV_LORA*V_DIM)/256, 256, 0, stream>>>(wkv_b, ws_WvT);
    // 8) flash attention over the latent cache (the 576 MB scan, TDM-fed)
    mla_flash_chunk<<<dim3(nchunk, BSZ), 512, fa_lds, stream>>>(
        ws_Q, kv_cache, pe_cache, ws_newkv, ws_pO, ws_pml, nchunk, chlen);
    // 9) LSE-combine partials
    mla_reduce<<<BSZ*RQ, 128, 0, stream>>>(ws_pO, ws_pml, ws_attn, nchunk);
    // 10) V projection per head: (128 x 512 x 128)
    wmma_gemm<<<dim3(V_DIM/64, NTOK/64, N_HEADS), 256, 0, stream>>>(
        ws_attn, N_HEADS*KV_LORA, KV_LORA,
        ws_WvT, V_DIM, (long long)KV_LORA*V_DIM,
        ws_out2, N_HEADS*V_DIM, V_DIM,
        NTOK, V_DIM, KV_LORA, 1.0f);
    // 11) final: out = out2 @ wo   (128 x 2048 x 2048)
    wmma_gemm<<<dim3(DIMM/64, NTOK/64, 1), 256, 0, stream>>>(
        ws_out2, N_HEADS*V_DIM, 0, wo, DIMM, 0, out, DIMM, 0,
        NTOK, DIMM, N_HEADS*V_DIM, 1.0f);
}